// S4D_77841987272900
// MI455X (gfx1250) — compile-verified
//
#include <hip/hip_runtime.h>
#include <cstdint>
#include <cstddef>

// ---- problem constants (match reference) ----
#define Hd   1024   // d_model
#define Nst  32     // d_state/2 complex modes  (== wave32 lanes)
#define Bb   4
#define Ld   4096
#define TILE 256    // timesteps per tile = 16 chunks x 16 steps
#define NCH  16     // chunks per tile
#define NT   (Ld / TILE)   // tiles per sequence = 16

typedef __attribute__((ext_vector_type(2))) float v2f;
typedef __attribute__((ext_vector_type(8))) float v8f;
typedef __attribute__((ext_vector_type(4))) unsigned int u32x4;
typedef __attribute__((ext_vector_type(8))) int i32x8;
typedef __attribute__((ext_vector_type(4))) int i32x4;

#if __has_builtin(__builtin_amdgcn_tensor_load_to_lds)
#define HAVE_TDM 1
#else
#define HAVE_TDM 0
#endif

// Issue one TDM descriptor: gather 256 H-strided f32 into LDS at lds_off.
__device__ __forceinline__ void tdm_gather_x(unsigned lds_off, unsigned long long ga)
{
#if HAVE_TDM
    u32x4 g0;
    g0[0] = 1u;                                  // count=1 valid descriptor
    g0[1] = lds_off;                             // lds_addr (bytes)
    g0[2] = (unsigned)ga;                        // global_addr[31:0]
    g0[3] = (unsigned)((ga >> 32) & 0x1FFFFFFull) | (2u << 30);  // addr[56:32] | type=2
    i32x8 g1;
    g1[0] = (int)(2u << 16);                     // data_size=4B; no multicast/pad/iterate
    g1[1] = (int)((unsigned)Hd << 16);           // tensor_dim0 = H (lo16)
    g1[2] = (int)((unsigned)TILE << 16);         // tensor_dim0 hi=0 | tensor_dim1=256 (lo16)
    g1[3] = (int)(1u << 16);                     // tensor_dim1 hi=0 | tile_dim0=1
    g1[4] = (int)TILE;                           // tile_dim1=256 | tile_dim2=0
    g1[5] = (int)Hd;                             // tensor_dim0_stride = H elements
    g1[6] = 0;
    g1[7] = 0;
    const i32x4 gz4 = {0, 0, 0, 0};
    const i32x8 gz8 = {0, 0, 0, 0, 0, 0, 0, 0};
    __builtin_amdgcn_tensor_load_to_lds(g0, g1, gz4, gz4, gz8, 0);
#endif
}

// One wave per (h, b). Lane n owns complex mode n.
__global__ __launch_bounds__(32) void s4d_scan_wmma(
    const float* __restrict__ x,         // [B, L, H]
    const float* __restrict__ log_dt,    // [H]
    const float* __restrict__ log_A_real,// [H, N]
    const float* __restrict__ A_imag,    // [H, N]
    const float* __restrict__ C,         // [H, N, 2]
    const float* __restrict__ Dp,        // [H]
    float* __restrict__ y)               // [B, L, H]
{
    const int h    = blockIdx.x;
    const int b    = blockIdx.y;
    const int lane = threadIdx.x;        // 0..31
    const int m    = lane & 15;          // row (A/C/D) or col (B) index
    const int hi   = lane >> 4;          // half-wave select

    __shared__ float sK[16];             // impulse response K[0..15]
    __shared__ float sT[16][16];         // intra-chunk lower-tri Toeplitz
    __shared__ float sG[16][64];         // inter-chunk output matrix (re/im interleaved cols)
    __shared__ float xbuf[2][TILE];      // double-buffered gathered x tiles
    __shared__ float Sbuf[NCH * 64];     // chunk-boundary states, [chunk][2n / 2n+1]

    // ---------------- per-mode parameters (lane n) ----------------
    const float dt  = __expf(log_dt[h]);
    const float Ar  = -__expf(log_A_real[h * Nst + lane]);
    const float Ai  = A_imag[h * Nst + lane];
    // w = exp(dt*A)
    const float er  = __expf(dt * Ar);
    const float wr  = er * __cosf(dt * Ai);
    const float wi  = er * __sinf(dt * Ai);
    // Cw = (C0 + i C1) * (w - 1) / A
    const float c0  = C[(h * Nst + lane) * 2 + 0];
    const float c1  = C[(h * Nst + lane) * 2 + 1];
    const float den = Ar * Ar + Ai * Ai;
    const float qr  = ((wr - 1.0f) * Ar + wi * Ai) / den;
    const float qi  = (wi * Ar - (wr - 1.0f) * Ai) / den;
    const float Cwr = c0 * qr - c1 * qi;
    const float Cwi = c0 * qi + c1 * qr;

    // w^16 by repeated squaring (chunk-to-chunk decay)
    float w16r = wr, w16i = wi;
    #pragma unroll
    for (int k = 0; k < 4; ++k) {
        const float a = w16r, bb2 = w16i;
        w16r = a * a - bb2 * bb2;
        w16i = 2.0f * a * bb2;
    }

    const float Dh = Dp[h];
    const size_t xbase = ((size_t)b * Ld) * Hd + h;
    const unsigned long long gbase =
        (unsigned long long)(uintptr_t)x + xbase * 4ull;
    const unsigned lds_x0 = (unsigned)(uintptr_t)(&xbuf[0][0]);
    const unsigned lds_x1 = (unsigned)(uintptr_t)(&xbuf[1][0]);

    // Kick off the gather for tile 0 before doing the table setup, so the
    // first tile's strided L2 gather overlaps the transcendental setup work.
    tdm_gather_x(lds_x0, gbase);

    // ---------------- build K[0..15] and G[16][64] ----------------
    {
        float pr = 1.0f, pi = 0.0f;      // w^d
        for (int d = 0; d < 16; ++d) {
            float r = 2.0f * (Cwr * pr - Cwi * pi);   // lane contribution to K[d]
            #pragma unroll
            for (int off = 16; off > 0; off >>= 1) r += __shfl_xor(r, off, 32);
            if (lane == 0) sK[d] = r;
            // advance to w^{d+1}; G[d][*] uses w^{d+1}
            const float npr = pr * wr - pi * wi;
            const float npi = pr * wi + pi * wr;
            pr = npr; pi = npi;
            sG[d][2 * lane + 0] =  2.0f * (Cwr * pr - Cwi * pi);
            sG[d][2 * lane + 1] = -2.0f * (Cwr * pi + Cwi * pr);
        }
    }
    // T[i][j] = (i >= j) ? K[i-j] : 0
    for (int idx = lane; idx < 256; idx += 32) {
        const int i = idx >> 4, j = idx & 15;
        sT[i][j] = (i >= j) ? sK[i - j] : 0.0f;
    }

    // Preload WMMA A-operands (f32 16x4 layout: lanes 0-15 K={4k,4k+1}, lanes 16-31 K={4k+2,4k+3})
    v2f aT[4];
    #pragma unroll
    for (int k = 0; k < 4; ++k) {
        aT[k].x = sT[m][4 * k + 2 * hi + 0];
        aT[k].y = sT[m][4 * k + 2 * hi + 1];
    }
    v2f aG[16];
    #pragma unroll
    for (int k = 0; k < 16; ++k) {
        aG[k].x = sG[m][4 * k + 2 * hi + 0];
        aG[k].y = sG[m][4 * k + 2 * hi + 1];
    }

    float Sr = 0.0f, Si = 0.0f;          // carried complex state (lane n)

    for (int t = 0; t < NT; ++t) {
        const float* __restrict__ xb = &xbuf[t & 1][0];

        // Prefetch next tile into the other buffer, then wait for this tile.
        // TENSORcnt completes in order within a wave.
        if (t + 1 < NT) {
            tdm_gather_x((t & 1) ? lds_x0 : lds_x1,
                         gbase + (unsigned long long)(t + 1) * TILE * Hd * 4ull);
#if HAVE_TDM
            __builtin_amdgcn_s_wait_tensorcnt(1);
#endif
        } else {
#if HAVE_TDM
            __builtin_amdgcn_s_wait_tensorcnt(0);
#endif
        }
#if !HAVE_TDM
        for (int idx = lane; idx < TILE; idx += 32)
            xbuf[t & 1][idx] = x[xbase + (size_t)(t * TILE + idx) * Hd];
#endif

        // ---- chunk-granular state recurrence; record chunk-entry states ----
        for (int c = 0; c < NCH; ++c) {
            Sbuf[c * 64 + 2 * lane + 0] = Sr;
            Sbuf[c * 64 + 2 * lane + 1] = Si;
            // dS = sum_j w^{15-j} x[c*16+j]  (Horner)
            float ar = 0.0f, ai2 = 0.0f;
            #pragma unroll
            for (int j = 0; j < 16; ++j) {
                const float xv = xb[c * 16 + j];
                const float na = ar * wr - ai2 * wi + xv;
                ai2 = ar * wi + ai2 * wr;
                ar = na;
            }
            const float nSr = w16r * Sr - w16i * Si + ar;
            const float nSi = w16i * Sr + w16r * Si + ai2;
            Sr = nSr; Si = nSi;
        }

        // ---- WMMA: Y(16x16) = T(16x16) @ X(16x16)  +  G(16x64) @ S(64x16) ----
        v8f acc = {};
        #pragma unroll
        for (int k = 0; k < 4; ++k) {
            v2f bb;                                      // B rows {4k+2hi, 4k+2hi+1}, col m
            bb.x = xb[m * 16 + 4 * k + 2 * hi + 0];
            bb.y = xb[m * 16 + 4 * k + 2 * hi + 1];
            acc = __builtin_amdgcn_wmma_f32_16x16x4_f32(
                false, aT[k], false, bb, (short)0, acc, false, false);
        }
        #pragma unroll
        for (int k = 0; k < 16; ++k) {
            v2f bb;
            bb.x = Sbuf[m * 64 + 4 * k + 2 * hi + 0];
            bb.y = Sbuf[m * 64 + 4 * k + 2 * hi + 1];
            acc = __builtin_amdgcn_wmma_f32_16x16x4_f32(
                false, aG[k], false, bb, (short)0, acc, false, false);
        }

        // ---- D-skip + strided store: lane holds col(chunk)=m, rows i=v+8*hi ----
        #pragma unroll
        for (int v = 0; v < 8; ++v) {
            const int i = v + 8 * hi;
            const int l = t * TILE + m * 16 + i;
            const float xv = xb[m * 16 + i];
            y[xbase + (size_t)l * Hd] = acc[v] + Dh * xv;
        }
    }
}

extern "C" void kernel_launch(void* const* d_in, const int* in_sizes, int n_in,
                              void* d_out, int out_size, void* d_ws, size_t ws_size,
                              hipStream_t stream) {
    const float* x          = (const float*)d_in[0];
    const float* log_dt     = (const float*)d_in[1];
    const float* log_A_real = (const float*)d_in[2];
    const float* A_imag     = (const float*)d_in[3];
    const float* C          = (const float*)d_in[4];
    const float* D          = (const float*)d_in[5];
    float* y = (float*)d_out;
    dim3 grid(Hd, Bb, 1);
    s4d_scan_wmma<<<grid, 32, 0, stream>>>(x, log_dt, log_A_real, A_imag, C, D, y);
}